// Group_48146583388932
// MI455X (gfx1250) — compile-verified
//
#include <hip/hip_runtime.h>
#include <hip/hip_bf16.h>

typedef __attribute__((ext_vector_type(2))) float v2f;
typedef __attribute__((ext_vector_type(8))) float v8f;

#define BATCH 32
#define NPTS  8192
#define NGRP  512
#define KNN   32
#define TILE  16      // centers per block (one WMMA tile)
#define CHUNK 512     // points per streamed distance tile
#define FPS_THREADS 1024
#define PTS_PER_THR (NPTS / FPS_THREADS)   // 8

// ---------------------------------------------------------------------------
// Kernel 1: farthest point sampling. One block per batch, whole point cloud in
// LDS (96 KB out of the WGP's 320 KB), min-distance state in registers.
// 511 serial iterations of: distance to last pick -> running min -> argmax.
// ---------------------------------------------------------------------------
__global__ __launch_bounds__(FPS_THREADS)
void fps_kernel(const float* __restrict__ xyz,
                float* __restrict__ center_out,     // [B,G,3]
                float* __restrict__ cidx_out) {     // [B,G] (as float)
    __shared__ float sx[NPTS];
    __shared__ float sy[NPTS];
    __shared__ float sz[NPTS];
    __shared__ float red_d[32];
    __shared__ int   red_i[32];
    __shared__ int   s_last;

    const int b    = blockIdx.x;
    const int tid  = threadIdx.x;
    const int w    = tid >> 5;
    const int lane = tid & 31;
    const float FINF = __builtin_inff();

    for (int p = tid; p < NPTS; p += FPS_THREADS) {
        sx[p] = xyz[(b * NPTS + p) * 3 + 0];
        sy[p] = xyz[(b * NPTS + p) * 3 + 1];
        sz[p] = xyz[(b * NPTS + p) * 3 + 2];
    }

    float mind[PTS_PER_THR];
#pragma unroll
    for (int j = 0; j < PTS_PER_THR; ++j) mind[j] = FINF;

    __syncthreads();

    if (tid == 0) {
        cidx_out[b * NGRP + 0] = 0.0f;
        center_out[(b * NGRP + 0) * 3 + 0] = sx[0];
        center_out[(b * NGRP + 0) * 3 + 1] = sy[0];
        center_out[(b * NGRP + 0) * 3 + 2] = sz[0];
    }

    int last = 0;
    for (int g = 1; g < NGRP; ++g) {
        const float lx = sx[last], ly = sy[last], lz = sz[last];
        float bd = -1.0f;
        int   bi = 0x7fffffff;
#pragma unroll
        for (int j = 0; j < PTS_PER_THR; ++j) {
            const int p = tid + j * FPS_THREADS;
            const float dx = sx[p] - lx;
            const float dy = sy[p] - ly;
            const float dz = sz[p] - lz;
            const float d = dx * dx + dy * dy + dz * dz;
            mind[j] = fminf(mind[j], d);
            const float md = mind[j];
            if (md > bd || (md == bd && p < bi)) { bd = md; bi = p; }
        }
        // wave32 argmax reduce (tie -> smaller index, matches jnp.argmax)
#pragma unroll
        for (int off = 16; off > 0; off >>= 1) {
            const float od = __shfl_xor(bd, off, 32);
            const int   oi = __shfl_xor(bi, off, 32);
            if (od > bd || (od == bd && oi < bi)) { bd = od; bi = oi; }
        }
        if (lane == 0) { red_d[w] = bd; red_i[w] = bi; }
        __syncthreads();
        if (w == 0) {
            float rd = red_d[lane];
            int   ri = red_i[lane];
#pragma unroll
            for (int off = 16; off > 0; off >>= 1) {
                const float od = __shfl_xor(rd, off, 32);
                const int   oi = __shfl_xor(ri, off, 32);
                if (od > rd || (od == rd && oi < ri)) { rd = od; ri = oi; }
            }
            if (lane == 0) {
                s_last = ri;
                cidx_out[b * NGRP + g] = (float)ri;
                center_out[(b * NGRP + g) * 3 + 0] = sx[ri];
                center_out[(b * NGRP + g) * 3 + 1] = sy[ri];
                center_out[(b * NGRP + g) * 3 + 2] = sz[ri];
            }
        }
        __syncthreads();
        last = s_last;
    }
}

// ---------------------------------------------------------------------------
// Kernel 2: kNN via fp32 WMMA (16x16x4) distance tiles + streaming top-32.
// Block = 8 waves, handles a tile of 16 centers. Each wave computes a 64-point
// slice of the 512-point d2 tile with WMMA into LDS, then owns 2 centers for
// candidate-list maintenance and the final 32-round wave-argmin merge.
// ---------------------------------------------------------------------------
__global__ __launch_bounds__(256)
void knn_kernel(const float* __restrict__ xyz,
                const float* __restrict__ center,   // [B,G,3]
                float* __restrict__ neigh,          // [B,G,K,3]
                float* __restrict__ idx_out) {      // [B,G,K] (as float)
    __shared__ float          s_d2[TILE][CHUNK];           // 32 KB
    __shared__ float          s_cd[TILE][32][KNN];         // 64 KB
    __shared__ unsigned short s_ci[TILE][32][KNN];         // 32 KB

    const int b    = blockIdx.y;
    const int g0   = blockIdx.x * TILE;
    const int tid  = threadIdx.x;
    const int w    = tid >> 5;
    const int lane = tid & 31;
    const int m    = lane & 15;
    const float FINF = __builtin_inff();

    for (int i = tid; i < TILE * 32 * KNN; i += 256) {
        (&s_cd[0][0][0])[i] = FINF;
        (&s_ci[0][0][0])[i] = 0;
    }

    // centers: lane L holds center m = L&15 (both halves, feeds A layout)
    const float cx = center[(b * NGRP + g0 + m) * 3 + 0];
    const float cy = center[(b * NGRP + g0 + m) * 3 + 1];
    const float cz = center[(b * NGRP + g0 + m) * 3 + 2];
    const float c2l = cx * cx + cy * cy + cz * cz;

    // A (16x4 fp32): VGPR0 = {K0 | K2}, VGPR1 = {K1 | K3} across lane halves
    v2f A;
    A.x = (lane < 16) ? cx : cz;
    A.y = (lane < 16) ? cy : 0.0f;

    // |c|^2 arranged in D layout: reg v holds row M = 8*(lane>>4)+v
    float c2v[8];
#pragma unroll
    for (int v = 0; v < 8; ++v)
        c2v[v] = __shfl(c2l, ((lane >> 4) << 3) + v, 32);

    float maxv[2]    = {FINF, FINF};
    int   maxslot[2] = {0, 0};

    __syncthreads();

    for (int chunk = 0; chunk < NPTS / CHUNK; ++chunk) {
        const int chunkBase = chunk * CHUNK;

        // ---- compute phase: wave w fills 64 columns of the d2 tile ----
#pragma unroll
        for (int it = 0; it < 4; ++it) {
            const int col = w * 64 + it * 16 + m;        // [0, CHUNK)
            const int p   = chunkBase + col;
            const float px = xyz[(b * NPTS + p) * 3 + 0];
            const float py = xyz[(b * NPTS + p) * 3 + 1];
            const float pz = xyz[(b * NPTS + p) * 3 + 2];
            const float p2 = px * px + py * py + pz * pz;

            v2f B;
            B.x = (lane < 16) ? px : pz;
            B.y = (lane < 16) ? py : 0.0f;

            v8f C = {};
            C = __builtin_amdgcn_wmma_f32_16x16x4_f32(
                    false, A, false, B, (short)0, C, false, false);

#pragma unroll
            for (int v = 0; v < 8; ++v) {
                const int mm = ((lane >> 4) << 3) + v;
                s_d2[mm][col] = c2v[v] + p2 - 2.0f * C[v];
            }
        }
        __syncthreads();

        // ---- insert phase: wave w owns centers 2w and 2w+1 ----
#pragma unroll
        for (int cc = 0; cc < 2; ++cc) {
            const int c = 2 * w + cc;
            float mv = maxv[cc];
            int   ms = maxslot[cc];
            for (int p = lane; p < CHUNK; p += 32) {
                const float d = s_d2[c][p];
                if (d < mv) {
                    s_cd[c][lane][ms] = d;
                    s_ci[c][lane][ms] = (unsigned short)(chunkBase + p);
                    mv = s_cd[c][lane][0];
                    ms = 0;
#pragma unroll
                    for (int s = 1; s < KNN; ++s) {
                        const float vv = s_cd[c][lane][s];
                        if (vv > mv) { mv = vv; ms = s; }
                    }
                }
            }
            maxv[cc] = mv;
            maxslot[cc] = ms;
        }
        __syncthreads();
    }

    // ---- merge phase: 32 rounds of wave-wide argmin -> ascending order ----
#pragma unroll 1
    for (int cc = 0; cc < 2; ++cc) {
        const int c = 2 * w + cc;
        const int g = g0 + c;
        int selIdx = 0;
        for (int k = 0; k < KNN; ++k) {
            float bv = FINF; int bs = 0; int bi = 0x7fffffff;
#pragma unroll
            for (int s = 0; s < KNN; ++s) {
                const float vv = s_cd[c][lane][s];
                const int   ii = (int)s_ci[c][lane][s];
                if (vv < bv || (vv == bv && ii < bi)) { bv = vv; bs = s; bi = ii; }
            }
            float rd = bv; int ri = bi;
#pragma unroll
            for (int off = 16; off > 0; off >>= 1) {
                const float od = __shfl_xor(rd, off, 32);
                const int   oi = __shfl_xor(ri, off, 32);
                if (od < rd || (od == rd && oi < ri)) { rd = od; ri = oi; }
            }
            if (bv == rd && bi == ri) s_cd[c][lane][bs] = FINF;  // retire winner
            if (lane == k) selIdx = ri;
        }
        // lane k writes neighbor k
        const float ccx = center[(b * NGRP + g) * 3 + 0];
        const float ccy = center[(b * NGRP + g) * 3 + 1];
        const float ccz = center[(b * NGRP + g) * 3 + 2];
        const float px = xyz[(b * NPTS + selIdx) * 3 + 0];
        const float py = xyz[(b * NPTS + selIdx) * 3 + 1];
        const float pz = xyz[(b * NPTS + selIdx) * 3 + 2];
        const long base = ((long)(b * NGRP + g)) * KNN + lane;
        neigh[base * 3 + 0] = px - ccx;
        neigh[base * 3 + 1] = py - ccy;
        neigh[base * 3 + 2] = pz - ccz;
        idx_out[base] = (float)selIdx;
    }
}

// ---------------------------------------------------------------------------
extern "C" void kernel_launch(void* const* d_in, const int* in_sizes, int n_in,
                              void* d_out, int out_size, void* d_ws, size_t ws_size,
                              hipStream_t stream) {
    const float* xyz = (const float*)d_in[0];
    float* out = (float*)d_out;

    const long n_neigh = (long)BATCH * NGRP * KNN * 3;   // 1,572,864
    const long n_cent  = (long)BATCH * NGRP * 3;         //    49,152
    const long n_idx   = (long)BATCH * NGRP * KNN;       //   524,288

    float* neigh  = out;
    float* centerO = out + n_neigh;
    float* idxO    = out + n_neigh + n_cent;
    float* cidxO   = out + n_neigh + n_cent + n_idx;

    fps_kernel<<<BATCH, FPS_THREADS, 0, stream>>>(xyz, centerO, cidxO);

    dim3 grid(NGRP / TILE, BATCH);   // (32, 32)
    knn_kernel<<<grid, 256, 0, stream>>>(xyz, centerO, neigh, idxO);
}